// GlobalGuidingModule_75136157876546
// MI455X (gfx1250) — compile-verified
//
#include <hip/hip_runtime.h>
#include <hip/hip_bf16.h>
#include <math.h>

// ---------------------------------------------------------------------------
// GlobalGuidingModule for MI455X (gfx1250, wave32, WMMA).
// Roofline: node_embeddings scan dominates (262 MB raw; ~50% skippable via
// wave-uniform mask => ~130 MB => ~6us @ 23.3 TB/s). GEMM compute ~2.6 GFLOP
// f32 -> microseconds on the matrix pipe; full-precision V_WMMA_F32_16X16X4.
// GEMM weights staged in LDS (64 KB/block, shared by 8 waves) with a swizzle
// that makes each lane's B fragment a contiguous b64.
// ---------------------------------------------------------------------------

typedef __attribute__((ext_vector_type(2))) float v2f;
typedef __attribute__((ext_vector_type(4))) float v4f;
typedef __attribute__((ext_vector_type(8))) float v8f;

#define GG_B 512
#define GG_N 1000
#define GG_C 50
#define GG_E 128
#define GG_NEG (-1000000000.0f)
#define GG_CLIP 10.0f

// ---------------------------------------------------------------------------
// Kernel 1: fused masked means over node_embeddings + context assembly.
// grid = B, block = 128. Thread t -> row-group (t>>5), columns 4*(t&31)..+3.
// Each wave owns one row-group => mask test is wave-uniform => masked rows are
// skipped entirely (halves HBM traffic) and loads are 16B per lane.
// ---------------------------------------------------------------------------
__global__ void gg_masked_means(const float* __restrict__ node,
                                const unsigned char* __restrict__ mask,
                                const unsigned char* __restrict__ cmask,
                                const float* __restrict__ current,
                                const float* __restrict__ depot,
                                float* __restrict__ ctx,      // B x 384
                                float* __restrict__ mean2) {  // B x 128
  const int b = blockIdx.x;
  const int t = threadIdx.x;  // 0..127
  const int rg = t >> 5;      // row group 0..3 (== wave id)
  const int cg = (t & 31) * 4;  // first of 4 owned columns
  __shared__ unsigned char sm[GG_N];
  __shared__ unsigned char sc[GG_N];
  __shared__ float r1[4][GG_E];
  __shared__ float r2[4][GG_E];
  for (int n = t; n < GG_N; n += 128) {
    sm[n] = mask[b * GG_N + n];
    sc[n] = cmask[b * GG_N + n];
  }
  __syncthreads();

  v4f s1, s2;
  for (int j = 0; j < 4; ++j) { s1[j] = 0.0f; s2[j] = 0.0f; }
  const float* np = node + (size_t)b * GG_N * GG_E + cg;
  for (int n = rg; n < GG_N; n += 4) {       // wave-uniform branch below
    if (!sm[n]) {
      const v4f x = *(const v4f*)(np + (size_t)n * GG_E);
      s1 += x;
      if (!sc[n]) s2 += x;
    }
  }
  for (int j = 0; j < 4; ++j) {
    r1[rg][cg + j] = s1[j];
    r2[rg][cg + j] = s2[j];
  }
  __syncthreads();

  // thread t reduces column t across the 4 row groups
  const float invn = 1.0f / (float)GG_N;
  const float m1 = (r1[0][t] + r1[1][t] + r1[2][t] + r1[3][t]) * invn;
  const float m2 = (r2[0][t] + r2[1][t] + r2[2][t] + r2[3][t]) * invn;
  ctx[(size_t)b * 384 + t]       = m1;
  ctx[(size_t)b * 384 + 128 + t] = current[(size_t)b * GG_E + t];
  ctx[(size_t)b * 384 + 256 + t] = depot[(size_t)b * GG_E + t];
  mean2[(size_t)b * GG_E + t]    = m2;
}

// ---------------------------------------------------------------------------
// WMMA f32 GEMM: C[M x 128] = A[M x K] @ B[K x 128], row-major,
// K % 128 == 0, M % 128 == 0 per grid (grid.x = M/128, block = 256 = 8 waves,
// each wave owns one 16-row strip; 8 v8f accumulators cover N=128).
//
// B is staged per 128-row K-chunk into LDS, swizzled so that the b64 pair
// (B[ka][n], B[ka+1][n]) a lane needs is contiguous:
//   sB_dword[ (r>>1)*256 + c*2 + (r&1) ] = B[kc + r][c]
// Fragment for k-step kk, half hi, col, tile t:
//   base_dw = (kk*2 + hi)*256 + col*2 + t*32  -> ds_load_b64
//
// A-frag (16x4 f32): lane L<16 -> M=L, {K0,K1}; lane L>=16 -> M=L-16, {K2,K3}.
// C/D (16x16): vgpr i, lanes 0-15 -> M=i, lanes 16-31 -> M=i+8.
// ---------------------------------------------------------------------------
__global__ void gg_wmma_gemm_n128(const float* __restrict__ A,
                                  const float* __restrict__ B,
                                  float* __restrict__ C,
                                  int K) {
  __shared__ float sB[128 * 128];  // 64 KB, shared by 8 waves
  const int tid  = threadIdx.x;       // 0..255
  const int wave = tid >> 5;          // 0..7
  const int lane = tid & 31;
  const int col  = lane & 15;
  const int hi   = lane >> 4;
  const int row0 = blockIdx.x * 128 + wave * 16;

  v8f acc[8];
#pragma unroll
  for (int t = 0; t < 8; ++t)
#pragma unroll
    for (int i = 0; i < 8; ++i) acc[t][i] = 0.0f;

  const float* arow = A + (size_t)(row0 + col) * K;

  for (int kc = 0; kc < K; kc += 128) {
    // ---- cooperative swizzled staging of B[kc..kc+127][0..127] ----
    __syncthreads();  // protect previous chunk's readers
    for (int lin = tid * 4; lin < 128 * 128; lin += 256 * 4) {
      const int r = lin >> 7;        // chunk row
      const int c = lin & 127;       // first of 4 consecutive cols
      const v4f vv = *(const v4f*)(B + (size_t)(kc + r) * 128 + c);
      float* dst = sB + (r >> 1) * 256 + (r & 1);
#pragma unroll
      for (int j = 0; j < 4; ++j) dst[(c + j) * 2] = vv[j];
    }
    __syncthreads();

    // ---- 32 k-steps over this chunk ----
    for (int kk = 0; kk < 32; ++kk) {
      const v2f a = *(const v2f*)(arow + kc + kk * 4 + 2 * hi);
      const float* bb = sB + (kk * 2 + hi) * 256 + col * 2;
#pragma unroll
      for (int t = 0; t < 8; ++t) {
        const v2f bf = *(const v2f*)(bb + t * 32);
        acc[t] = __builtin_amdgcn_wmma_f32_16x16x4_f32(
            /*neg_a=*/false, a, /*neg_b=*/false, bf,
            /*c_mod=*/(short)0, acc[t], /*reuse_a=*/false, /*reuse_b=*/false);
      }
    }
  }

#pragma unroll
  for (int t = 0; t < 8; ++t)
#pragma unroll
    for (int i = 0; i < 8; ++i)
      C[(size_t)(row0 + i + 8 * hi) * 128 + t * 16 + col] = acc[t][i];
}

// ---------------------------------------------------------------------------
// Kernel 3: per-batch multi-head attention (8 heads x 50 keys, q-len 1).
// grid = B, block = 128.
// ---------------------------------------------------------------------------
__global__ void gg_attention(const float* __restrict__ qh,   // B x 128
                             const float* __restrict__ kh,   // B*50 x 128
                             const float* __restrict__ vh,   // B*50 x 128
                             const unsigned char* __restrict__ visited,  // B x 50
                             float* __restrict__ glimpse_h) {            // B x 128
  const int b = blockIdx.x;
  const int t = threadIdx.x;  // 0..127
  __shared__ float q_s[128];
  __shared__ float sc[8][GG_C];
  __shared__ unsigned char vcm_s[GG_C];

  q_s[t] = qh[(size_t)b * 128 + t];
  if (t == 0) {
    bool all = true;
    for (int j = 1; j < GG_C; ++j) all = all && (visited[b * GG_C + j] != 0);
    vcm_s[0] = all ? 0 : 1;  // ~all_visited
  } else if (t < GG_C) {
    vcm_s[t] = visited[b * GG_C + t];
  }
  __syncthreads();

  for (int p = t; p < 8 * GG_C; p += 128) {
    const int h = p / GG_C;
    const int j = p % GG_C;
    if (vcm_s[j]) {
      sc[h][j] = GG_NEG;
    } else {
      const float* kp = kh + ((size_t)b * GG_C + j) * 128 + h * 16;
      float s = 0.0f;
      for (int d = 0; d < 16; ++d) s += q_s[h * 16 + d] * kp[d];
      sc[h][j] = s * 0.25f;  // 1/sqrt(16)
    }
  }
  __syncthreads();

  if (t < 8) {
    float mx = -INFINITY;
    for (int j = 0; j < GG_C; ++j) mx = fmaxf(mx, sc[t][j]);
    float sum = 0.0f;
    for (int j = 0; j < GG_C; ++j) {
      float e = expf(sc[t][j] - mx);
      sc[t][j] = e;
      sum += e;
    }
    float inv = 1.0f / sum;
    for (int j = 0; j < GG_C; ++j) sc[t][j] *= inv;
  }
  __syncthreads();

  const int h = t >> 4;
  float g = 0.0f;
  for (int j = 0; j < GG_C; ++j)
    g += sc[h][j] * vh[((size_t)b * GG_C + j) * 128 + t];
  glimpse_h[(size_t)b * 128 + t] = g;
}

// ---------------------------------------------------------------------------
// Kernel 5: logits, tanh clip, masked log-softmax, argmax, output selection.
// grid = B, block = 128.
// ---------------------------------------------------------------------------
__global__ void gg_finalize(const float* __restrict__ glimpse,  // B x 128
                            const float* __restrict__ pk,       // B*50 x 128
                            const unsigned char* __restrict__ visited,
                            const unsigned char* __restrict__ is_new,
                            const int* __restrict__ guidance_in,
                            const float* __restrict__ cluster_emb,  // B*50 x 128
                            const float* __restrict__ guid_emb_in,  // B x 128
                            const float* __restrict__ aug_in,       // B x 512
                            const float* __restrict__ mean2,        // B x 128
                            const float* __restrict__ current,
                            const float* __restrict__ depot,
                            float* __restrict__ out_aug,      // B x 512
                            float* __restrict__ out_gembed,   // B x 128
                            float* __restrict__ out_guid,     // B
                            float* __restrict__ out_cluprob)  // B x 50
{
  const int b = blockIdx.x;
  const int t = threadIdx.x;
  __shared__ float g_s[128];
  __shared__ float logit[GG_C];
  __shared__ unsigned char vcm_s[GG_C];
  __shared__ float lse_s;
  __shared__ int arg_s;

  g_s[t] = glimpse[(size_t)b * 128 + t];
  if (t == 0) {
    bool all = true;
    for (int j = 1; j < GG_C; ++j) all = all && (visited[b * GG_C + j] != 0);
    vcm_s[0] = all ? 0 : 1;
  } else if (t < GG_C) {
    vcm_s[t] = visited[b * GG_C + t];
  }
  __syncthreads();

  if (t < GG_C) {
    if (vcm_s[t]) {
      logit[t] = GG_NEG;
    } else {
      const float* pp = pk + ((size_t)b * GG_C + t) * 128;
      float s = 0.0f;
      for (int e = 0; e < 128; ++e) s += g_s[e] * pp[e];
      s *= 0.08838834764831845f;  // 1/sqrt(128)
      logit[t] = tanhf(s) * GG_CLIP;
    }
  }
  __syncthreads();

  if (t == 0) {
    float mx = -INFINITY;
    int arg = 0;
    for (int c = 0; c < GG_C; ++c)
      if (logit[c] > mx) { mx = logit[c]; arg = c; }  // first max, like argmax
    float sum = 0.0f;
    for (int c = 0; c < GG_C; ++c) sum += expf(logit[c] - mx);
    lse_s = mx + logf(sum);
    arg_s = arg;
  }
  __syncthreads();

  const bool nw = is_new[b] != 0;
  const int g = arg_s;
  const float lse = lse_s;

  for (int c = t; c < GG_C; c += 128)
    out_cluprob[(size_t)b * GG_C + c] = nw ? (logit[c] - lse) : 0.0f;
  if (t == 0)
    out_guid[b] = (float)(nw ? g : guidance_in[b]);

  const float ge = cluster_emb[((size_t)b * GG_C + g) * 128 + t];
  out_gembed[(size_t)b * 128 + t] =
      nw ? ge : guid_emb_in[(size_t)b * 128 + t];

  const float* ai = aug_in + (size_t)b * 512;
  float* ao = out_aug + (size_t)b * 512;
  ao[t]       = nw ? mean2[(size_t)b * 128 + t]   : ai[t];
  ao[128 + t] = nw ? current[(size_t)b * 128 + t] : ai[128 + t];
  ao[256 + t] = nw ? ge                            : ai[256 + t];
  ao[384 + t] = nw ? depot[(size_t)b * 128 + t]   : ai[384 + t];
}

// ---------------------------------------------------------------------------
extern "C" void kernel_launch(void* const* d_in, const int* in_sizes, int n_in,
                              void* d_out, int out_size, void* d_ws, size_t ws_size,
                              hipStream_t stream) {
  (void)in_sizes; (void)n_in; (void)out_size; (void)ws_size;

  const float* depot      = (const float*)d_in[0];   // B x 128
  const float* cluster    = (const float*)d_in[1];   // B x 50 x 128
  const float* current    = (const float*)d_in[2];   // B x 128
  const float* node       = (const float*)d_in[3];   // B x 1000 x 128
  const float* aug_ctx    = (const float*)d_in[4];   // B x 512
  const float* guid_emb   = (const float*)d_in[5];   // B x 128
  const float* Wq         = (const float*)d_in[6];   // 384 x 128
  const float* Wk         = (const float*)d_in[7];   // 128 x 128
  const float* Wv         = (const float*)d_in[8];   // 128 x 128
  const float* Wks        = (const float*)d_in[9];   // 128 x 128
  const float* mha_Wq     = (const float*)d_in[10];  // 128 x 128
  const float* mha_Wk     = (const float*)d_in[11];
  const float* mha_Wv     = (const float*)d_in[12];
  const float* mha_Wo     = (const float*)d_in[13];
  const unsigned char* is_new   = (const unsigned char*)d_in[14];  // B
  const unsigned char* mask     = (const unsigned char*)d_in[15];  // B x 1000
  const unsigned char* cmask    = (const unsigned char*)d_in[16];  // B x 1000
  const unsigned char* visited  = (const unsigned char*)d_in[17];  // B x 50
  const int* guidance_in        = (const int*)d_in[18];            // B
  // d_in[19] = step (unused)

  // ---- workspace layout (floats) ----
  float* ws = (float*)d_ws;
  size_t o = 0;
  float* ctx    = ws + o; o += (size_t)GG_B * 384;        // context (B x 384)
  float* mean2  = ws + o; o += (size_t)GG_B * 128;        // unvisit_cluster mean
  float* Wqp    = ws + o; o += (size_t)384 * 128;         // Wq @ mha_Wq
  float* Wkp    = ws + o; o += (size_t)128 * 128;         // Wk @ mha_Wk
  float* Wvp    = ws + o; o += (size_t)128 * 128;         // Wv @ mha_Wv
  float* qh     = ws + o; o += (size_t)GG_B * 128;
  float* kh     = ws + o; o += (size_t)GG_B * GG_C * 128;
  float* vh     = ws + o; o += (size_t)GG_B * GG_C * 128;
  float* pk     = ws + o; o += (size_t)GG_B * GG_C * 128;
  float* gh     = ws + o; o += (size_t)GG_B * 128;        // head-concat glimpse
  float* gl     = ws + o; o += (size_t)GG_B * 128;        // glimpse @ Wo

  // ---- output layout (floats, tuple concatenated flat) ----
  float* out         = (float*)d_out;
  float* out_aug     = out;                               // B x 512
  float* out_gembed  = out_aug + (size_t)GG_B * 512;      // B x 128
  float* out_guid    = out_gembed + (size_t)GG_B * 128;   // B
  float* out_cluprob = out_guid + GG_B;                   // B x 50

  // 1. masked means + context assembly (bandwidth-dominant pass)
  gg_masked_means<<<GG_B, 128, 0, stream>>>(node, mask, cmask, current, depot,
                                            ctx, mean2);

  // 2. fused weights: W' = W @ mha_W  (projection + head-split fused)
  gg_wmma_gemm_n128<<<3, 256, 0, stream>>>(Wq, mha_Wq, Wqp, 128);   // M=384
  gg_wmma_gemm_n128<<<1, 256, 0, stream>>>(Wk, mha_Wk, Wkp, 128);   // M=128
  gg_wmma_gemm_n128<<<1, 256, 0, stream>>>(Wv, mha_Wv, Wvp, 128);   // M=128

  // 3. batched projections
  gg_wmma_gemm_n128<<<GG_B / 128, 256, 0, stream>>>(ctx, Wqp, qh, 384);
  gg_wmma_gemm_n128<<<(GG_B * GG_C) / 128, 256, 0, stream>>>(cluster, Wkp, kh, 128);
  gg_wmma_gemm_n128<<<(GG_B * GG_C) / 128, 256, 0, stream>>>(cluster, Wvp, vh, 128);
  gg_wmma_gemm_n128<<<(GG_B * GG_C) / 128, 256, 0, stream>>>(cluster, Wks, pk, 128);

  // 4. attention (8 heads x 50 keys per batch)
  gg_attention<<<GG_B, 128, 0, stream>>>(qh, kh, vh, visited, gh);

  // 5. output projection
  gg_wmma_gemm_n128<<<GG_B / 128, 256, 0, stream>>>(gh, mha_Wo, gl, 128);

  // 6. logits, log-softmax, argmax, gathered/selected outputs
  gg_finalize<<<GG_B, 128, 0, stream>>>(gl, pk, visited, is_new, guidance_in,
                                        cluster, guid_emb, aug_ctx, mean2,
                                        current, depot, out_aug, out_gembed,
                                        out_guid, out_cluprob);
}